// SimpleGCN_42030549959108
// MI455X (gfx1250) — compile-verified
//
#include <hip/hip_runtime.h>
#include <hip/hip_bf16.h>

typedef __attribute__((ext_vector_type(2))) float v2f;
typedef __attribute__((ext_vector_type(8))) float v8f;

#define F_IN   128
#define F_HID  256
#define F_OUT  128
#define HCAP   (1u << 20)          // hash capacity (E = 512K -> load factor 0.5)
#define HMASK  (HCAP - 1u)
#define HEMPTY 0xFFFFFFFFu

// ---------------------------------------------------------------------------
// Init: agg1 = x (self loop, weight 1), degree = 1
// ---------------------------------------------------------------------------
__global__ void gcn_init_self(const float* __restrict__ x,
                              float* __restrict__ agg1,
                              float* __restrict__ degree, int n) {
    int idx = blockIdx.x * blockDim.x + threadIdx.x;
    if (idx < n) degree[idx] = 1.0f;
    if (idx < n * F_IN) agg1[idx] = x[idx];
}

// ---------------------------------------------------------------------------
// Hash insert: last-write-wins dedupe. Key = row*N + col, value = edge_id+1.
// ---------------------------------------------------------------------------
__global__ void gcn_hash_insert(const int* __restrict__ ei,
                                unsigned int* __restrict__ hkeys,
                                unsigned int* __restrict__ hvals,
                                int nedges, int n) {
    int e = blockIdx.x * blockDim.x + threadIdx.x;
    if (e >= nedges) return;
    unsigned int key = (unsigned int)ei[e] * (unsigned int)n + (unsigned int)ei[nedges + e];
    unsigned int slot = (key * 2654435761u) & HMASK;
    for (;;) {
        unsigned int prev = atomicCAS(&hkeys[slot], HEMPTY, key);
        if (prev == HEMPTY || prev == key) {
            atomicMax(&hvals[slot], (unsigned int)(e + 1));
            break;
        }
        slot = (slot + 1u) & HMASK;
    }
}

// ---------------------------------------------------------------------------
// Edge aggregation, 128 features: one wave32 per edge, 4 floats per lane.
// Winning edge (per last-write-wins) contributes w * x[col] into agg[row],
// and w into degree[row].
// ---------------------------------------------------------------------------
__global__ void __launch_bounds__(256)
gcn_agg_edges_128(const int* __restrict__ ei, const float* __restrict__ ew,
                  const unsigned int* __restrict__ hkeys,
                  const unsigned int* __restrict__ hvals,
                  const float* __restrict__ X, float* __restrict__ agg,
                  float* __restrict__ degree, int nedges, int n) {
    int e = blockIdx.x * 8 + (threadIdx.x >> 5);
    if (e >= nedges) return;
    int lane = threadIdx.x & 31;
    int row = ei[e];
    int col = ei[nedges + e];
    unsigned int key = (unsigned int)row * (unsigned int)n + (unsigned int)col;
    unsigned int slot = (key * 2654435761u) & HMASK;
    while (hkeys[slot] != key) slot = (slot + 1u) & HMASK;
    if (hvals[slot] != (unsigned int)(e + 1)) return;   // superseded duplicate
    float w = ew[e];
    if (lane == 0) atomicAdd(degree + row, w);
    const float4 xv = *(const float4*)(X + (size_t)col * F_IN + lane * 4);
    float* dst = agg + (size_t)row * F_IN + lane * 4;
    atomicAdd(dst + 0, w * xv.x);
    atomicAdd(dst + 1, w * xv.y);
    atomicAdd(dst + 2, w * xv.z);
    atomicAdd(dst + 3, w * xv.w);
}

// Same, 256 features: 8 floats per lane.
__global__ void __launch_bounds__(256)
gcn_agg_edges_256(const int* __restrict__ ei, const float* __restrict__ ew,
                  const unsigned int* __restrict__ hkeys,
                  const unsigned int* __restrict__ hvals,
                  const float* __restrict__ H, float* __restrict__ agg,
                  int nedges, int n) {
    int e = blockIdx.x * 8 + (threadIdx.x >> 5);
    if (e >= nedges) return;
    int lane = threadIdx.x & 31;
    int row = ei[e];
    int col = ei[nedges + e];
    unsigned int key = (unsigned int)row * (unsigned int)n + (unsigned int)col;
    unsigned int slot = (key * 2654435761u) & HMASK;
    while (hkeys[slot] != key) slot = (slot + 1u) & HMASK;
    if (hvals[slot] != (unsigned int)(e + 1)) return;
    float w = ew[e];
    const float* src = H + (size_t)col * F_HID + lane * 8;
    float* dst = agg + (size_t)row * F_HID + lane * 8;
    const float4 a = *(const float4*)(src);
    const float4 b = *(const float4*)(src + 4);
    atomicAdd(dst + 0, w * a.x);
    atomicAdd(dst + 1, w * a.y);
    atomicAdd(dst + 2, w * a.z);
    atomicAdd(dst + 3, w * a.w);
    atomicAdd(dst + 4, w * b.x);
    atomicAdd(dst + 5, w * b.y);
    atomicAdd(dst + 6, w * b.z);
    atomicAdd(dst + 7, w * b.w);
}

// ---------------------------------------------------------------------------
// Row normalization: A[i][f] /= max(degree[i], 1)
// ---------------------------------------------------------------------------
__global__ void gcn_norm_rows(float* __restrict__ A,
                              const float* __restrict__ degree,
                              int total, int f) {
    int idx = blockIdx.x * blockDim.x + threadIdx.x;
    if (idx >= total) return;
    float d = fmaxf(degree[idx / f], 1.0f);
    A[idx] = A[idx] / d;
}

// ---------------------------------------------------------------------------
// GEMM + bias via V_WMMA_F32_16X16X4_F32 (fp32, matches reference precision).
// Block = 128 threads = 4 waves; one 16-row panel per block, 4 column tiles.
// A panel [16 x K] staged in LDS (stride K+4 -> conflict-free ds loads).
// VGPR layouts per CDNA5 ISA 7.12.2:
//   A 16x4:  lanes 0-15 M=0..15; v0 = K=0 / K=2 (halves), v1 = K=1 / K=3
//   B 4x16:  lanes hold N; v0 = K row 0 / 2 (halves), v1 = K row 1 / 3
//   D 16x16: VGPR r -> M = r (lanes 0-15), M = r+8 (lanes 16-31), N = lane&15
// ---------------------------------------------------------------------------
template <int K, int NCOLS>
__global__ void __launch_bounds__(128)
gcn_gemm_bias_wmma(const float* __restrict__ A, const float* __restrict__ B,
                   const float* __restrict__ bias, float* __restrict__ D) {
    __shared__ float smem[16 * (K + 4)];
    const int LDS_STRIDE = K + 4;

    const int tm   = blockIdx.x * 16;
    const int wave = threadIdx.x >> 5;
    const int lane = threadIdx.x & 31;
    const int tn   = (blockIdx.y * 4 + wave) * 16;

    // Cooperative load of the A panel into LDS.
    #pragma unroll
    for (int i = threadIdx.x; i < 16 * K; i += 128) {
        int r = i / K;
        int c = i % K;
        smem[r * LDS_STRIDE + c] = A[(size_t)(tm + r) * K + c];
    }
    __syncthreads();

    const int half = lane >> 4;   // which K pair this half-wave holds
    const int lid  = lane & 15;

    v8f acc;
    const float bv = bias[tn + lid];
    #pragma unroll
    for (int r = 0; r < 8; ++r) acc[r] = bv;

    const float* Ap = smem + lid * LDS_STRIDE + 2 * half;
    const float* Bp = B + (size_t)(2 * half) * NCOLS + tn + lid;

    #pragma unroll 4
    for (int k = 0; k < K; k += 4) {
        v2f a;
        a.x = Ap[k];
        a.y = Ap[k + 1];
        v2f b;
        b.x = Bp[0];
        b.y = Bp[NCOLS];
        acc = __builtin_amdgcn_wmma_f32_16x16x4_f32(
            /*neg_a=*/false, a, /*neg_b=*/false, b,
            /*c_mod=*/(short)0, acc, /*reuse_a=*/false, /*reuse_b=*/false);
        Bp += 4 * NCOLS;
    }

    float* Dp = D + (size_t)(tm + half * 8) * NCOLS + tn + lid;
    #pragma unroll
    for (int r = 0; r < 8; ++r) Dp[(size_t)r * NCOLS] = acc[r];
}

// ---------------------------------------------------------------------------
// LayerNorm + ReLU over F_HID=256, one wave32 per row (8 elems/lane).
// Writes normalized h AND initializes agg2 with it (self-loop, weight 1).
// ---------------------------------------------------------------------------
__global__ void __launch_bounds__(256)
gcn_ln_relu(const float* __restrict__ Hin, const float* __restrict__ g,
            const float* __restrict__ b, float* __restrict__ Hout,
            float* __restrict__ agg2, int n) {
    int row = blockIdx.x * 8 + (threadIdx.x >> 5);
    if (row >= n) return;
    int lane = threadIdx.x & 31;
    const float* rp = Hin + (size_t)row * F_HID + lane * 8;

    float v[8];
    {
        float4 p0 = *(const float4*)(rp);
        float4 p1 = *(const float4*)(rp + 4);
        v[0] = p0.x; v[1] = p0.y; v[2] = p0.z; v[3] = p0.w;
        v[4] = p1.x; v[5] = p1.y; v[6] = p1.z; v[7] = p1.w;
    }
    float s = 0.f, ss = 0.f;
    #pragma unroll
    for (int r = 0; r < 8; ++r) { s += v[r]; ss += v[r] * v[r]; }
    #pragma unroll
    for (int m = 16; m >= 1; m >>= 1) {
        s  += __shfl_xor(s,  m, 32);
        ss += __shfl_xor(ss, m, 32);
    }
    const float mu   = s * (1.0f / F_HID);
    const float var  = ss * (1.0f / F_HID) - mu * mu;
    const float rstd = rsqrtf(var + 1e-5f);

    #pragma unroll
    for (int r = 0; r < 8; ++r) {
        int f = lane * 8 + r;
        float o = (v[r] - mu) * rstd * g[f] + b[f];
        v[r] = fmaxf(o, 0.0f);
    }
    float* op = Hout + (size_t)row * F_HID + lane * 8;
    float* ap = agg2 + (size_t)row * F_HID + lane * 8;
    float4 q0 = make_float4(v[0], v[1], v[2], v[3]);
    float4 q1 = make_float4(v[4], v[5], v[6], v[7]);
    *(float4*)(op)     = q0;
    *(float4*)(op + 4) = q1;
    *(float4*)(ap)     = q0;
    *(float4*)(ap + 4) = q1;
}

// ---------------------------------------------------------------------------
// Host-side launcher
// ---------------------------------------------------------------------------
extern "C" void kernel_launch(void* const* d_in, const int* in_sizes, int n_in,
                              void* d_out, int out_size, void* d_ws, size_t ws_size,
                              hipStream_t stream) {
    const float* x    = (const float*)d_in[0];
    const int*   ei   = (const int*)d_in[1];
    const float* ew   = (const float*)d_in[2];
    const float* W1   = (const float*)d_in[3];
    const float* b1   = (const float*)d_in[4];
    const float* W2   = (const float*)d_in[5];
    const float* b2   = (const float*)d_in[6];
    const float* ln_g = (const float*)d_in[7];
    const float* ln_b = (const float*)d_in[8];
    float* out = (float*)d_out;

    const int N = in_sizes[0] / F_IN;     // 16384
    const int E = in_sizes[2];            // 524288

    // Workspace carve-up
    char* ws = (char*)d_ws;
    unsigned int* hkeys  = (unsigned int*)ws;                    ws += (size_t)HCAP * 4;
    unsigned int* hvals  = (unsigned int*)ws;                    ws += (size_t)HCAP * 4;
    float*        degree = (float*)ws;                           ws += (size_t)N * 4;
    float*        agg1   = (float*)ws;                           ws += (size_t)N * F_IN * 4;
    float*        hbuf   = (float*)ws;                           ws += (size_t)N * F_HID * 4;
    float*        agg2   = (float*)ws;                           ws += (size_t)N * F_HID * 4;

    // Hash table init (graph-capture-safe async memsets)
    hipMemsetAsync(hkeys, 0xFF, (size_t)HCAP * 4, stream);
    hipMemsetAsync(hvals, 0x00, (size_t)HCAP * 4, stream);

    // agg1 = x, degree = 1
    {
        int total = N * F_IN;
        gcn_init_self<<<(total + 255) / 256, 256, 0, stream>>>(x, agg1, degree, N);
    }
    // last-write-wins dedupe
    gcn_hash_insert<<<(E + 255) / 256, 256, 0, stream>>>(ei, hkeys, hvals, E, N);
    // edge aggregation layer 1 (+degree)
    gcn_agg_edges_128<<<(E + 7) / 8, 256, 0, stream>>>(ei, ew, hkeys, hvals,
                                                       x, agg1, degree, E, N);
    // normalize
    gcn_norm_rows<<<(N * F_IN + 255) / 256, 256, 0, stream>>>(agg1, degree, N * F_IN, F_IN);
    // hbuf = agg1 @ W1 + b1   [N,128]x[128,256]
    {
        dim3 grid(N / 16, F_HID / 64);
        gcn_gemm_bias_wmma<F_IN, F_HID><<<grid, 128, 0, stream>>>(agg1, W1, b1, hbuf);
    }
    // LN + ReLU; also seeds agg2 with the self-loop contribution
    gcn_ln_relu<<<(N + 7) / 8, 256, 0, stream>>>(hbuf, ln_g, ln_b, hbuf, agg2, N);
    // edge aggregation layer 2
    gcn_agg_edges_256<<<(E + 7) / 8, 256, 0, stream>>>(ei, ew, hkeys, hvals,
                                                       hbuf, agg2, E, N);
    // normalize
    gcn_norm_rows<<<(N * F_HID + 255) / 256, 256, 0, stream>>>(agg2, degree, N * F_HID, F_HID);
    // out = agg2 @ W2 + b2   [N,256]x[256,128]
    {
        dim3 grid(N / 16, F_OUT / 64);
        gcn_gemm_bias_wmma<F_HID, F_OUT><<<grid, 128, 0, stream>>>(agg2, W2, b2, out);
    }
}